// CustomQuantumLayer_16097537425540
// MI455X (gfx1250) — compile-verified
//
#include <hip/hip_runtime.h>
#include <hip/hip_bf16.h>

// MI455X (gfx1250) quantum-circuit batch simulator.
//
// Math: out[b,q] = sum_n sign_q(n) * |(U @ st0_b)[n]|^2 where
//   st0_b = real product state from RY data encoding (16 floats / sample),
//   U     = fixed 16x16 complex unitary from the two entangling layers.
// Per 16-sample tile: 3 chained-K GEMMs on v_wmma_f32_16x16x4_f32
// (Yr, Yi, and the signed reduction), plus elementwise square in VALU.

typedef __attribute__((ext_vector_type(2))) float v2f;
typedef __attribute__((ext_vector_type(8))) float v8f;

#define WAVES_PER_BLOCK 8

// ---------------------------------------------------------------------------
// Setup: build U (16x16 complex) from weights[2][4][3]. Thread t owns column t.
// Stored in ws as: ws[n*16+k] = Re U[n][k], ws[256+n*16+k] = Im U[n][k].
// (That row-major layout IS the B-matrix layout B[k][n] = U[n][k] we need.)
// ---------------------------------------------------------------------------
__global__ void build_unitary_kernel(const float* __restrict__ w,
                                     float* __restrict__ U) {
  const int t = threadIdx.x;  // 0..15, column of U
  float vr[16], vi[16];
#pragma unroll
  for (int n = 0; n < 16; ++n) { vr[n] = (n == t) ? 1.f : 0.f; vi[n] = 0.f; }

#pragma unroll
  for (int l = 0; l < 2; ++l) {
    // Rot(phi, theta, omega) = RZ(omega) RY(theta) RZ(phi) on each qubit
#pragma unroll
    for (int q = 0; q < 4; ++q) {
      const float phi = w[(l * 4 + q) * 3 + 0];
      const float th  = w[(l * 4 + q) * 3 + 1];
      const float om  = w[(l * 4 + q) * 3 + 2];
      const float a = 0.5f * (phi + om);
      const float b = 0.5f * (phi - om);
      float st, ct, sa, ca, sb, cb;
      __sincosf(0.5f * th, &st, &ct);
      __sincosf(a, &sa, &ca);
      __sincosf(b, &sb, &cb);
      // g00 = e^{-ia} ct ; g01 = -e^{ib} st ; g10 = e^{-ib} st ; g11 = e^{ia} ct
      const float g00r =  ca * ct, g00i = -sa * ct;
      const float g01r = -cb * st, g01i = -sb * st;
      const float g10r =  cb * st, g10i = -sb * st;
      const float g11r =  ca * ct, g11i =  sa * ct;
      const int m = 1 << (3 - q);  // qubit q -> amplitude bit (3-q)
#pragma unroll
      for (int n = 0; n < 16; ++n) {
        if (!(n & m)) {
          const int n1 = n | m;
          const float xr = vr[n],  xi = vi[n];
          const float yr = vr[n1], yi = vi[n1];
          vr[n]  = g00r * xr - g00i * xi + g01r * yr - g01i * yi;
          vi[n]  = g00r * xi + g00i * xr + g01r * yi + g01i * yr;
          vr[n1] = g10r * xr - g10i * xi + g11r * yr - g11i * yi;
          vi[n1] = g10r * xi + g10i * xr + g11r * yi + g11i * yr;
        }
      }
    }
    // CNOT ring, range r = l+1: CNOT(q, (q+r)%4) -> swap rows with control set
    const int r = l + 1;
#pragma unroll
    for (int q = 0; q < 4; ++q) {
      const int tq = (q + r) & 3;
      const int cm = 1 << (3 - q);
      const int tm = 1 << (3 - tq);
#pragma unroll
      for (int n = 0; n < 16; ++n) {
        if ((n & cm) && !(n & tm)) {
          const int n1 = n | tm;
          float tmp;
          tmp = vr[n]; vr[n] = vr[n1]; vr[n1] = tmp;
          tmp = vi[n]; vi[n] = vi[n1]; vi[n1] = tmp;
        }
      }
    }
  }
#pragma unroll
  for (int n = 0; n < 16; ++n) {
    U[n * 16 + t]       = vr[n];
    U[256 + n * 16 + t] = vi[n];
  }
}

// ---------------------------------------------------------------------------
// Main kernel: one wave handles 16 samples per tile via three WMMA chains.
// f32 16x16x4 A-layout: lane L -> M = L&15; K-block j: v0 = [M][4j+2*hi],
// v1 = [M][4j+2*hi+1] (hi = L>=16). B-layout mirrors with N = L&15.
// ---------------------------------------------------------------------------
__global__ __launch_bounds__(32 * WAVES_PER_BLOCK)
void qcircuit_kernel(const float* __restrict__ x, const float* __restrict__ U,
                     float* __restrict__ out, int ntiles) {
  __shared__ float stage[WAVES_PER_BLOCK][256];

  const int lane = threadIdx.x & 31;
  const int wave = threadIdx.x >> 5;
  const int hi   = lane >> 4;   // which half-wave (upper K pair)
  const int col  = lane & 15;   // M for A / N for B / N for D

  // ---- loop-invariant B fragments: Ur^T, Ui^T (from ws) and sign matrix S
  v2f Br[4], Bi[4], Bs[4];
  const float* Urow = U + col * 16;        // Re U[n=col][k]
  const float* Virow = U + 256 + col * 16; // Im U[n=col][k]
#pragma unroll
  for (int j = 0; j < 4; ++j) {
    const int k = 4 * j + 2 * hi;
    Br[j] = *(const v2f*)(Urow + k);   // k even -> 8B aligned
    Bi[j] = *(const v2f*)(Virow + k);
    // S[k][q]: q = col (<4), amplitude bit for qubit q is bit (3-q)
    float s0 = 0.f, s1 = 0.f;
    if (col < 4) {
      s0 = (((unsigned)k >> (3 - col)) & 1u) ? -1.f : 1.f;
      s1 = ((((unsigned)k + 1u) >> (3 - col)) & 1u) ? -1.f : 1.f;
    }
    Bs[j][0] = s0;
    Bs[j][1] = s1;
  }

  const int waveId = blockIdx.x * WAVES_PER_BLOCK + wave;
  const int nWaves = gridDim.x * WAVES_PER_BLOCK;

  for (int tile = waveId; tile < ntiles; tile += nWaves) {
    const int samp = tile * 16 + col;  // sample this lane builds (M = col)
    const float4 xv = ((const float4*)x)[samp];

    // st0 factors: f_q(0)=cos(x_q/2), f_q(1)=sin(x_q/2)
    float c0, s0, c1, s1, c2, s2, c3, s3;
    __sincosf(0.5f * xv.x, &s0, &c0);
    __sincosf(0.5f * xv.y, &s1, &c1);
    __sincosf(0.5f * xv.z, &s2, &c2);
    __sincosf(0.5f * xv.w, &s3, &c3);

    // K index bits of k = 4j + 2*hi + e: i0=j>>1, i1=j&1, i2=hi, i3=e
    const float u2 = hi ? s2 : c2;
    const float wc = u2 * c3, ws = u2 * s3;
    const float p0 = c0 * c1, p1 = c0 * s1, p2 = s0 * c1, p3 = s0 * s1;
    v2f A[4];
    A[0][0] = p0 * wc; A[0][1] = p0 * ws;
    A[1][0] = p1 * wc; A[1][1] = p1 * ws;
    A[2][0] = p2 * wc; A[2][1] = p2 * ws;
    A[3][0] = p3 * wc; A[3][1] = p3 * ws;

    // Yr = ST0 * Ur^T ; Yi = ST0 * Ui^T  (K=16 via 4 chained 16x16x4 WMMAs)
    v8f yr = {};
    v8f yi = {};
#pragma unroll
    for (int j = 0; j < 4; ++j) {
      yr = __builtin_amdgcn_wmma_f32_16x16x4_f32(false, A[j], false, Br[j],
                                                 (short)0, yr, false, false);
      yi = __builtin_amdgcn_wmma_f32_16x16x4_f32(false, A[j], false, Bi[j],
                                                 (short)0, yi, false, false);
    }

    // probabilities P = Yr^2 + Yi^2 (D-layout: VGPR r -> M=r+8*hi, N=col)
    const v8f p = yr * yr + yi * yi;

    // D-layout -> A-layout transpose of P through per-wave LDS tile
#pragma unroll
    for (int r2 = 0; r2 < 8; ++r2)
      stage[wave][(r2 + 8 * hi) * 16 + col] = p[r2];
    asm volatile("s_wait_dscnt 0" ::: "memory");  // intra-wave LDS RAW

    v2f PA[4];
#pragma unroll
    for (int j = 0; j < 4; ++j) {
      const int k = 4 * j + 2 * hi;
      PA[j][0] = stage[wave][col * 16 + k];
      PA[j][1] = stage[wave][col * 16 + k + 1];
    }

    // EV = P * S  (signed reduction as a third WMMA chain)
    v8f ev = {};
#pragma unroll
    for (int j = 0; j < 4; ++j)
      ev = __builtin_amdgcn_wmma_f32_16x16x4_f32(false, PA[j], false, Bs[j],
                                                 (short)0, ev, false, false);

    // D[m][q]: lanes with col<4 hold valid q columns (EXEC divergence only
    // here, after all WMMAs of this tile)
    if (col < 4) {
#pragma unroll
      for (int r2 = 0; r2 < 8; ++r2) {
        const int m = r2 + 8 * hi;
        out[(tile * 16 + m) * 4 + col] = ev[r2];
      }
    }
  }
}

// ---------------------------------------------------------------------------
extern "C" void kernel_launch(void* const* d_in, const int* in_sizes, int n_in,
                              void* d_out, int out_size, void* d_ws,
                              size_t ws_size, hipStream_t stream) {
  const float* x = (const float*)d_in[0];      // [B,4] f32
  const float* w = (const float*)d_in[1];      // [2,4,3] f32
  float* out = (float*)d_out;                  // [B,4] f32
  float* U = (float*)d_ws;                     // 512 floats: Re|Im of 16x16 U

  const int B = in_sizes[0] / 4;
  const int ntiles = B / 16;

  build_unitary_kernel<<<dim3(1), dim3(16), 0, stream>>>(w, U);

  const int blocks = 1024;  // 8192 waves -> ~8 tiles per wave at B=1M
  qcircuit_kernel<<<dim3(blocks), dim3(32 * WAVES_PER_BLOCK), 0, stream>>>(
      x, U, out, ntiles);
}